// ProcNetLoss_86526411145916
// MI455X (gfx1250) — compile-verified
//
#include <hip/hip_runtime.h>
#include <math.h>

// Problem constants from the reference
#define KK   64
#define LL   4096
#define NN   128
#define CC   4096
#define KL   (KK * LL)            // 262144 proposals
#define BLK  256
#define PPT  4                    // proposals per thread (register blocking)
#define NBLK_A (KL / (BLK * PPT)) // 256 blocks in the main kernel

// Workspace layout (floats):
//   [0,    256)  per-block sum of masked BCE
//   [256,  512)  per-block sum of mask
//   [512,  768)  per-block sum of smooth-L1 numerator
//   [768,  896)  per-row  -log p[gt]  (sequential loss)
#define WS_BCE  0
#define WS_MSK  (NBLK_A)
#define WS_SL1  (2 * NBLK_A)
#define WS_SEQ  (3 * NBLK_A)

__device__ __forceinline__ float block_reduce_sum(float v, float* red, int tid) {
  red[tid] = v;
  __syncthreads();
#pragma unroll
  for (int s = BLK / 2; s > 0; s >>= 1) {
    if (tid < s) red[tid] += red[tid + s];
    __syncthreads();
  }
  float r = red[0];
  __syncthreads();
  return r;
}

// ---------------------------------------------------------------------------
// Kernel A: IoU matching + masked BCE + smooth-L1.
// GT table (2*NN = 256 floats) is broadcast into LDS with the CDNA5 async
// LDS-DMA path (GLOBAL_LOAD_ASYNC_TO_LDS_B32, tracked by ASYNCcnt), then
// repacked into (start,end) float2 pairs so the hot loop is one ds_load_b64
// per GT segment feeding PPT=4 register-blocked proposals (division-free IoU).
// ---------------------------------------------------------------------------
__global__ __launch_bounds__(BLK) void loss_main_kernel(
    const float* __restrict__ scores,    // (K,L)
    const float* __restrict__ boundary,  // (2,K,L): low at [idx], high at [KL+idx]
    const float* __restrict__ gt,        // (2,N)
    float* __restrict__ ws) {
  __shared__ float  s_raw[2 * NN];   // async-loaded as-is: starts then ends
  __shared__ float2 s_pair[NN];      // interleaved (g0,g1) pairs
  __shared__ float  red[BLK];

  const int tid = threadIdx.x;

  // Async copy gt -> LDS: one B32 per lane (256 lanes == 256 floats).
  {
    const float* gp = gt + tid;
    // Low 32 bits of the generic pointer to an LDS object are the LDS offset
    // (ISA aperture rule).
    unsigned lds_addr = (unsigned)(size_t)(&s_raw[tid]);
    asm volatile("global_load_async_to_lds_b32 %0, %1, off"
                 :: "v"(lds_addr), "v"(gp)
                 : "memory");
    asm volatile("s_wait_asynccnt 0" ::: "memory");
  }
  __syncthreads();   // all waves' async loads complete + visible

  if (tid < NN) {
    s_pair[tid] = make_float2(s_raw[tid], s_raw[NN + tid]);
  }
  __syncthreads();

  // Load PPT proposals per thread (each j-slice is a coalesced stride-BLK load).
  const int base = blockIdx.x * (BLK * PPT);
  float lo[PPT], hi[PPT], p[PPT];
#pragma unroll
  for (int j = 0; j < PPT; ++j) {
    const int idx = base + j * BLK + tid;
    lo[j] = boundary[idx];
    hi[j] = boundary[KL + idx];
    p[j]  = scores[idx];
  }

  unsigned posM = 0, negGtM = 0;        // bit j: any iou>0.8 / any iou>0.2
  float bi[PPT], bu[PPT];               // best inter/union (iou = bi/bu)
  int   bidx[PPT];
#pragma unroll
  for (int j = 0; j < PPT; ++j) { bi[j] = 0.0f; bu[j] = 1.0f; bidx[j] = -1; }

#pragma unroll 2
  for (int n = 0; n < NN; ++n) {
    const float2 g = s_pair[n];         // one ds_load_b64 feeds 4 proposals
#pragma unroll
    for (int j = 0; j < PPT; ++j) {
      const float inter = fminf(hi[j], g.y) - fmaxf(lo[j], g.x);
      const float uni   = fmaxf(hi[j], g.y) - fminf(lo[j], g.x);  // > 0 always
      const bool hasI   = inter > 0.0f;
      if (hasI && (inter > 0.2f * uni)) negGtM |= (1u << j);      // iou > NEG_T
      if (hasI && (inter > 0.8f * uni)) {                         // iou > POS_T
        posM |= (1u << j);
        // inter/uni > bi/bu  <=>  inter*bu > bi*uni  (first-max argmax)
        if (bidx[j] < 0 || inter * bu[j] > bi[j] * uni) {
          bi[j] = inter; bu[j] = uni; bidx[j] = n;
        }
      }
    }
  }

  float bce = 0.0f, maskv = 0.0f, sl1 = 0.0f;
#pragma unroll
  for (int j = 0; j < PPT; ++j) {
    const bool valid = (lo[j] > 0.0f) && (hi[j] <= (float)LL);
    const bool pos   = valid && ((posM >> j) & 1u);
    const bool neg   = valid && !pos && !((negGtM >> j) & 1u);
    if (pos || neg) {
      maskv += 1.0f;
      const float lp = pos ? fmaxf(logf(p[j]), -100.0f)
                           : fmaxf(log1pf(-p[j]), -100.0f);
      bce -= lp;
    }
    if (pos) {
      const float2 gb = s_pair[bidx[j]];
      const float d0 = fabsf(lo[j] - gb.x);
      const float d1 = fabsf(hi[j] - gb.y);
      sl1 += (d0 < 1.0f ? 0.5f * d0 * d0 : d0 - 0.5f) +
             (d1 < 1.0f ? 0.5f * d1 * d1 : d1 - 0.5f);
    }
  }

  const float sb = block_reduce_sum(bce,   red, tid);
  const float sm = block_reduce_sum(maskv, red, tid);
  const float ss = block_reduce_sum(sl1,   red, tid);
  if (tid == 0) {
    ws[WS_BCE + blockIdx.x] = sb;
    ws[WS_MSK + blockIdx.x] = sm;
    ws[WS_SL1 + blockIdx.x] = ss;
  }
}

// ---------------------------------------------------------------------------
// Kernel B: per-row log-softmax + gather. One block per row (N=128 rows,
// C=4096 cols). max-reduce, exp-sum-reduce, then -logp at the gt index.
// ---------------------------------------------------------------------------
__global__ __launch_bounds__(BLK) void seq_loss_kernel(
    const float* __restrict__ X,      // (N,C)
    const int*   __restrict__ enc,    // (N,)
    float* __restrict__ ws) {
  __shared__ float red[BLK];
  const int row = blockIdx.x;
  const int tid = threadIdx.x;
  const float* xr = X + (size_t)row * CC;

  float m = -3.402823466e+38f;
  for (int c = tid; c < CC; c += BLK) m = fmaxf(m, xr[c]);
  red[tid] = m;
  __syncthreads();
#pragma unroll
  for (int s = BLK / 2; s > 0; s >>= 1) {
    if (tid < s) red[tid] = fmaxf(red[tid], red[tid + s]);
    __syncthreads();
  }
  m = red[0];
  __syncthreads();

  float sum = 0.0f;
  for (int c = tid; c < CC; c += BLK) sum += expf(xr[c] - m);
  sum = block_reduce_sum(sum, red, tid);

  if (tid == 0) {
    const float lse = m + logf(sum);
    const int g = enc[row];
    ws[WS_SEQ + row] = lse - xr[g];   // == -log_softmax(x)[g]
  }
}

// ---------------------------------------------------------------------------
// Kernel C: deterministic finalize — reduce all partials, emit the scalar.
// ---------------------------------------------------------------------------
__global__ __launch_bounds__(BLK) void finalize_kernel(
    const float* __restrict__ ws, float* __restrict__ out) {
  __shared__ float red[BLK];
  const int tid = threadIdx.x;

  float a = 0.0f, b = 0.0f, c = 0.0f, d = 0.0f;
  if (tid < NBLK_A) {                 // NBLK_A == 256 == BLK
    a = ws[WS_BCE + tid];
    b = ws[WS_MSK + tid];
    c = ws[WS_SL1 + tid];
  }
  if (tid < NN) d = ws[WS_SEQ + tid];

  a = block_reduce_sum(a, red, tid);
  b = block_reduce_sum(b, red, tid);
  c = block_reduce_sum(c, red, tid);
  d = block_reduce_sum(d, red, tid);

  if (tid == 0) {
    const float loss_cls = a / b;
    const float loss_bnd = c * (1.0f / (float)(2 * KL));
    const float loss_seq = d * (1.0f / (float)NN);
    out[0] = loss_cls + loss_bnd + loss_seq;
  }
}

extern "C" void kernel_launch(void* const* d_in, const int* in_sizes, int n_in,
                              void* d_out, int out_size, void* d_ws, size_t ws_size,
                              hipStream_t stream) {
  const float* scores   = (const float*)d_in[0];  // (1,K,L)
  const float* boundary = (const float*)d_in[1];  // (1,2,K,L)
  const int*   enc      = (const int*)  d_in[2];  // (N,)
  const float* seqidx   = (const float*)d_in[3];  // (N,C)
  const float* gt       = (const float*)d_in[4];  // (1,2,N)
  float* ws  = (float*)d_ws;
  float* out = (float*)d_out;

  loss_main_kernel<<<NBLK_A, BLK, 0, stream>>>(scores, boundary, gt, ws);
  seq_loss_kernel<<<NN, BLK, 0, stream>>>(seqidx, enc, ws);
  finalize_kernel<<<1, BLK, 0, stream>>>(ws, out);
}